// CentralMambaBlock_88167088652525
// MI455X (gfx1250) — compile-verified
//
#include <hip/hip_runtime.h>
#include <math.h>

// Problem constants (from reference)
#define B_    2
#define NCH_  32
#define IC_   64
#define S_    16
#define R_    4
#define NB_   200
#define NCS_  8
#define L_    7
#define NPIX_ 56            // NCS_*L_
#define CPD_  11200         // NPIX_*NB_ : per-channel plane stride
#define BN_   16            // B_*NCS_
#define MTOT_ 22400         // B_*NPIX_*NB_ == BN_*L_*NB_

typedef __attribute__((ext_vector_type(2))) float v2f;
typedef __attribute__((ext_vector_type(8))) float v8f;

__device__ __forceinline__ v8f wmma4(v2f a, v2f b, v8f c) {
#if defined(__gfx1250__)
  // V_WMMA_F32_16X16X4_F32 : D = A(16x4) x B(4x16) + C, all f32, wave32
  return __builtin_amdgcn_wmma_f32_16x16x4_f32(false, a, false, b, (short)0, c, false, false);
#else
  (void)a; (void)b; return c;
#endif
}

__device__ __forceinline__ float swishf(float v)   { return v / (1.f + __expf(-v)); }
__device__ __forceinline__ float softplusf(float z){ return z > 20.f ? z : log1pf(__expf(z)); }

// ---------------------------------------------------------------------------
// Kernel 1: x_and_res = W_in^T x + b_in   (M=22400, K=32, N=128) via WMMA f32
// x: (B,32,NPIX,NB)  ->  xr: (B,128,NPIX,NB)
// ---------------------------------------------------------------------------
__global__ void k_pw_in(const float* __restrict__ x, const float* __restrict__ W_in,
                        const float* __restrict__ b_in, float* __restrict__ xr) {
  int wid = blockIdx.x * 8 + (threadIdx.x >> 5);
  if (wid >= MTOT_ / 16) return;                 // wave-uniform
  int lane = threadIdx.x & 31;
  int mr = lane & 15, kh = lane >> 4, nl = lane & 15;
  int m0 = wid * 16;
  int m = m0 + mr;
  int d = m % NB_; int bp = m / NB_; int p = bp % NPIX_; int b = bp / NPIX_;
  const float* xa = x + ((size_t)b * NCH_) * CPD_ + p * NB_ + d;
  v2f a[8];
  #pragma unroll
  for (int kk = 0; kk < 8; ++kk) {
    int c0 = kk * 4 + kh * 2;
    a[kk].x = xa[(size_t)c0 * CPD_];
    a[kk].y = xa[(size_t)(c0 + 1) * CPD_];
  }
  for (int nt = 0; nt < 8; ++nt) {
    int n = nt * 16 + nl;
    float bias = b_in[n];
    v8f acc; for (int r = 0; r < 8; ++r) acc[r] = bias;
    #pragma unroll
    for (int kk = 0; kk < 8; ++kk) {
      int c0 = kk * 4 + kh * 2;
      v2f bm; bm.x = W_in[c0 * 128 + n]; bm.y = W_in[(c0 + 1) * 128 + n];
      acc = wmma4(a[kk], bm, acc);
    }
    #pragma unroll
    for (int r = 0; r < 8; ++r) {
      int mm = m0 + r + 8 * kh;
      int dd = mm % NB_; int bp2 = mm / NB_; int pp = bp2 % NPIX_; int bb = bp2 / NPIX_;
      xr[(((size_t)bb * 128 + n) * NPIX_ + pp) * NB_ + dd] = acc[r];
    }
  }
}

// ---------------------------------------------------------------------------
// Kernel 2: xc = conv7(xm[:, :, p=0, :]) + b_cc ;  E = xc . W_xcp   (tiny)
// block = 64 channels x 4 d ; grid = B * 50
// ---------------------------------------------------------------------------
__global__ void k_conv_xc(const float* __restrict__ xr, const float* __restrict__ W_cc,
                          const float* __restrict__ b_cc, const float* __restrict__ W_xcp,
                          float* __restrict__ xc, float* __restrict__ Eb) {
  int o = threadIdx.x & 63, dq = threadIdx.x >> 6;
  int b = blockIdx.x / 50, d = (blockIdx.x % 50) * 4 + dq;
  float acc = b_cc[o];
  for (int k = 0; k < 7; ++k) {
    int dd = d + k - 3;
    if (dd < 0 || dd >= NB_) continue;
    const float* xin = xr + ((size_t)b * 128) * CPD_ + dd;   // p == 0 plane
    const float* w = W_cc + (k * IC_) * IC_ + o;
    for (int c = 0; c < IC_; ++c) acc += w[c * IC_] * xin[(size_t)c * CPD_];
  }
  xc[((size_t)b * IC_ + o) * NB_ + d] = acc;
  __shared__ float sxc[4][IC_];
  sxc[dq][o] = acc;
  __syncthreads();
  if (o < S_) {
    float e = 0.f;
    for (int c = 0; c < IC_; ++c) e += sxc[dq][c] * W_xcp[c * S_ + o];
    Eb[((size_t)b * S_ + o) * NB_ + d] = e;
  }
}

// ---------------------------------------------------------------------------
// Kernel 3: xs = swish(conv7(xm per (bn,l) row) + b_cs)  via WMMA
// GEMM view: M = d rows (16 per tile), N = 64 out-ch, K = 7 taps x 64 ch
// xs layout: (BN,64,L,NB)
// ---------------------------------------------------------------------------
__global__ void k_conv_xs(const float* __restrict__ xr, const float* __restrict__ W_cs,
                          const float* __restrict__ b_cs, float* __restrict__ xs) {
  int wid = blockIdx.x * 8 + (threadIdx.x >> 5);
  if (wid >= BN_ * L_ * 13) return;              // 1456 wave tiles, uniform
  int lane = threadIdx.x & 31, mr = lane & 15, kh = lane >> 4, nl = lane & 15;
  int dt = wid % 13, bnl = wid / 13;
  int l = bnl % L_, bn = bnl / L_;
  int b = bn >> 3, ncs = bn & 7, p = ncs * L_ + l;
  int d0 = dt * 16, d = d0 + mr;
  const float* xin = xr + ((size_t)b * 128) * CPD_ + p * NB_;
  v8f acc[4];
  for (int nt = 0; nt < 4; ++nt) { float bb = b_cs[nt * 16 + nl]; for (int r = 0; r < 8; ++r) acc[nt][r] = bb; }
  for (int tap = 0; tap < 7; ++tap) {
    int dd = d + tap - 3;
    bool va = (d < NB_) && dd >= 0 && dd < NB_;
    const float* wt = W_cs + tap * IC_ * IC_;
    #pragma unroll 4
    for (int cs = 0; cs < 16; ++cs) {
      int c0 = cs * 4 + kh * 2;
      v2f a;
      a.x = va ? xin[(size_t)c0 * CPD_ + dd] : 0.f;
      a.y = va ? xin[(size_t)(c0 + 1) * CPD_ + dd] : 0.f;
      for (int nt = 0; nt < 4; ++nt) {
        int n = nt * 16 + nl;
        v2f bm; bm.x = wt[c0 * IC_ + n]; bm.y = wt[(c0 + 1) * IC_ + n];
        acc[nt] = wmma4(a, bm, acc[nt]);
      }
    }
  }
  for (int nt = 0; nt < 4; ++nt) {
    int n = nt * 16 + nl;
    for (int r = 0; r < 8; ++r) {
      int ds = d0 + r + 8 * kh;
      if (ds < NB_)
        xs[(((size_t)bn * IC_ + n) * L_ + l) * NB_ + ds] = swishf(acc[nt][r]);
    }
  }
}

// ---------------------------------------------------------------------------
// Kernel 4: x_dbc = xs . W_xp  (M=22400, K=64, N=36) via WMMA, split stores:
// dp (BN,L,NB,4), Bm/Cm (BN,L,NB,16)
// ---------------------------------------------------------------------------
__global__ void k_pw_dbc(const float* __restrict__ xs, const float* __restrict__ W_xp,
                         float* __restrict__ dp, float* __restrict__ Bmb, float* __restrict__ Cmb) {
  int wid = blockIdx.x * 8 + (threadIdx.x >> 5);
  if (wid >= MTOT_ / 16) return;
  int lane = threadIdx.x & 31, mr = lane & 15, kh = lane >> 4, nl = lane & 15;
  int m0 = wid * 16, m = m0 + mr;
  int d = m % NB_; int bl = m / NB_; int l = bl % L_; int bn = bl / L_;
  const float* xa = xs + (((size_t)bn * IC_) * L_ + l) * NB_ + d;   // + c*(L_*NB_)
  v8f acc[3];
  for (int nt = 0; nt < 3; ++nt) for (int r = 0; r < 8; ++r) acc[nt][r] = 0.f;
  #pragma unroll 4
  for (int cs = 0; cs < 16; ++cs) {
    int c0 = cs * 4 + kh * 2;
    v2f a; a.x = xa[(size_t)c0 * (L_ * NB_)]; a.y = xa[(size_t)(c0 + 1) * (L_ * NB_)];
    for (int nt = 0; nt < 3; ++nt) {
      int j = nt * 16 + nl;
      v2f bm;
      bm.x = (j < 36) ? W_xp[c0 * 36 + j] : 0.f;
      bm.y = (j < 36) ? W_xp[(c0 + 1) * 36 + j] : 0.f;
      acc[nt] = wmma4(a, bm, acc[nt]);
    }
  }
  for (int nt = 0; nt < 3; ++nt) {
    int j = nt * 16 + nl;
    for (int r = 0; r < 8; ++r) {
      int mm = m0 + r + 8 * kh;
      float v = acc[nt][r];
      if (j < R_)            dp[(size_t)mm * R_ + j] = v;
      else if (j < R_ + S_)  Bmb[(size_t)mm * S_ + (j - R_)] = v;
      else if (j < 36)       Cmb[(size_t)mm * S_ + (j - 20)] = v;
    }
  }
}

// ---------------------------------------------------------------------------
// Block scan helpers (224 threads = 7 waves, wave32 shuffles + LDS carries)
// ---------------------------------------------------------------------------
__device__ __forceinline__ void block_rev_ex_scan(float* vs, float* segTot, int lane, int wv) {
  float incl[S_];
  #pragma unroll
  for (int s = 0; s < S_; ++s) {
    float x = vs[s];
    #pragma unroll
    for (int off = 1; off < 32; off <<= 1) {
      float t = __shfl_down(x, off, 32);
      if (lane + off < 32) x += t;
    }
    incl[s] = x;                         // reverse-inclusive within wave
  }
  __syncthreads();                       // protect segTot from previous use
  if (lane == 0) {
    for (int s = 0; s < S_; ++s) segTot[wv * S_ + s] = incl[s];
  }
  __syncthreads();
  #pragma unroll
  for (int s = 0; s < S_; ++s) {
    float add = 0.f;
    for (int w2 = wv + 1; w2 < 7; ++w2) add += segTot[w2 * S_ + s];
    vs[s] = incl[s] + add - vs[s];       // reverse-exclusive across block
  }
}

__device__ __forceinline__ void block_fwd_in_scan(float* vs, float* segTot, int lane, int wv) {
  float incl[S_];
  #pragma unroll
  for (int s = 0; s < S_; ++s) {
    float x = vs[s];
    #pragma unroll
    for (int off = 1; off < 32; off <<= 1) {
      float t = __shfl_up(x, off, 32);
      if (lane >= off) x += t;
    }
    incl[s] = x;                         // forward-inclusive within wave
  }
  __syncthreads();
  if (lane == 31) {
    for (int s = 0; s < S_; ++s) segTot[wv * S_ + s] = incl[s];
  }
  __syncthreads();
  #pragma unroll
  for (int s = 0; s < S_; ++s) {
    float add = 0.f;
    for (int w2 = 0; w2 < wv; ++w2) add += segTot[w2 * S_ + s];
    vs[s] = incl[s] + add;               // forward-inclusive across block
  }
}

// ---------------------------------------------------------------------------
// Kernel 5: fused 2-D SSM scan. One block per (bn,c); thread = column d,
// s in registers.  P[l,d] = T[d] - sum_{a<=l} rowSuffix_a[d] avoids any
// materialization of dAc/h (92 MB in the reference path).
// Also applies delta=softplus(dp.W_dt+b_dt), y gate with swish(res).
// ym layout: (B,64,NPIX,NB)
// ---------------------------------------------------------------------------
__global__ void __launch_bounds__(224) k_scan(
    const float* __restrict__ xr, const float* __restrict__ xc, const float* __restrict__ Eb,
    const float* __restrict__ xs, const float* __restrict__ dp,
    const float* __restrict__ Bmb, const float* __restrict__ Cmb,
    const float* __restrict__ W_dt, const float* __restrict__ b_dt,
    const float* __restrict__ A_log, const float* __restrict__ Dvec,
    float* __restrict__ ym) {
  __shared__ float segTot[7 * S_];
  int c = blockIdx.x & 63, bn = blockIdx.x >> 6, b = bn >> 3;
  int tid = threadIdx.x, lane = tid & 31, wv = tid >> 5;
  int d = tid; bool act = d < NB_;

  float Arow[S_];
  #pragma unroll
  for (int s = 0; s < S_; ++s) Arow[s] = -__expf(A_log[c * S_ + s]);
  float wdt0 = W_dt[c], wdt1 = W_dt[64 + c], wdt2 = W_dt[128 + c], wdt3 = W_dt[192 + c];
  float bdt = b_dt[c], Dc = Dvec[c];
  float xcd = act ? xc[((size_t)b * IC_ + c) * NB_ + d] : 0.f;
  float Es[S_];
  #pragma unroll
  for (int s = 0; s < S_; ++s) Es[s] = act ? Eb[((size_t)b * S_ + s) * NB_ + d] : 0.f;

  // Phase 1: T[d][s] = sum over ALL rows a of (reverse-exclusive suffix of dA row a)
  float T[S_];
  #pragma unroll
  for (int s = 0; s < S_; ++s) T[s] = 0.f;
  for (int l = 0; l < L_; ++l) {
    float dl = 0.f;
    if (act) {
      const float* q = dp + (((size_t)bn * L_ + l) * NB_ + d) * R_;
      dl = softplusf(bdt + q[0] * wdt0 + q[1] * wdt1 + q[2] * wdt2 + q[3] * wdt3);
    }
    float vs[S_];
    #pragma unroll
    for (int s = 0; s < S_; ++s) vs[s] = dl * Arow[s];
    block_rev_ex_scan(vs, segTot, lane, wv);
    #pragma unroll
    for (int s = 0; s < S_; ++s) T[s] += vs[s];
  }

  // Phase 2: forward over l: P = T - Pacc ; 2-D prefix numerator ; y
  float Pacc[S_], Nacc[S_];
  #pragma unroll
  for (int s = 0; s < S_; ++s) { Pacc[s] = 0.f; Nacc[s] = 0.f; }
  for (int l = 0; l < L_; ++l) {
    float dl = 0.f;
    if (act) {
      const float* q = dp + (((size_t)bn * L_ + l) * NB_ + d) * R_;
      dl = softplusf(bdt + q[0] * wdt0 + q[1] * wdt1 + q[2] * wdt2 + q[3] * wdt3);
    }
    float xsv = act ? xs[(((size_t)bn * IC_ + c) * L_ + l) * NB_ + d] : 0.f;
    size_t mb = (((size_t)bn * L_ + l) * NB_ + d) * S_;
    float vs[S_];
    #pragma unroll
    for (int s = 0; s < S_; ++s) vs[s] = dl * Arow[s];
    block_rev_ex_scan(vs, segTot, lane, wv);
    float eP[S_], ws[S_];
    #pragma unroll
    for (int s = 0; s < S_; ++s) {
      Pacc[s] += vs[s];                          // sum_{a<=l} rowSuffix_a[d]
      eP[s] = __expf(T[s] - Pacc[s]);            // dAc[l][d]
      float Brs = act ? Bmb[mb + s] : 0.f;
      ws[s] = dl * (xsv * Brs + xcd * Es[s]) * eP[s];
    }
    block_fwd_in_scan(ws, segTot, lane, wv);
    float yv = xsv * Dc;
    #pragma unroll
    for (int s = 0; s < S_; ++s) {
      Nacc[s] += ws[s];                          // 2-D prefix numerator
      float Crs = act ? Cmb[mb + s] : 0.f;
      yv += (Nacc[s] / (eP[s] + 1e-12f)) * Crs;
    }
    if (act) {
      int p = (bn & 7) * L_ + l;
      float rv = xr[(((size_t)b * 128 + IC_ + c) * NPIX_ + p) * NB_ + d];
      ym[(((size_t)b * IC_ + c) * NPIX_ + p) * NB_ + d] = yv * swishf(rv);
    }
  }
}

// ---------------------------------------------------------------------------
// Kernel 6: out = ym . W_out + b_out   (M=22400, K=64, N=32) via WMMA
// ---------------------------------------------------------------------------
__global__ void k_pw_out(const float* __restrict__ ym, const float* __restrict__ W_out,
                         const float* __restrict__ b_out, float* __restrict__ out) {
  int wid = blockIdx.x * 8 + (threadIdx.x >> 5);
  if (wid >= MTOT_ / 16) return;
  int lane = threadIdx.x & 31, mr = lane & 15, kh = lane >> 4, nl = lane & 15;
  int m0 = wid * 16, m = m0 + mr;
  int d = m % NB_; int bp = m / NB_; int p = bp % NPIX_; int b = bp / NPIX_;
  const float* xa = ym + ((size_t)b * IC_) * CPD_ + p * NB_ + d;
  v8f acc[2];
  for (int nt = 0; nt < 2; ++nt) { float bb = b_out[nt * 16 + nl]; for (int r = 0; r < 8; ++r) acc[nt][r] = bb; }
  #pragma unroll 4
  for (int cs = 0; cs < 16; ++cs) {
    int c0 = cs * 4 + kh * 2;
    v2f a; a.x = xa[(size_t)c0 * CPD_]; a.y = xa[(size_t)(c0 + 1) * CPD_];
    for (int nt = 0; nt < 2; ++nt) {
      int n = nt * 16 + nl;
      v2f bm; bm.x = W_out[c0 * NCH_ + n]; bm.y = W_out[(c0 + 1) * NCH_ + n];
      acc[nt] = wmma4(a, bm, acc[nt]);
    }
  }
  for (int nt = 0; nt < 2; ++nt) {
    int n = nt * 16 + nl;
    for (int r = 0; r < 8; ++r) {
      int mm = m0 + r + 8 * kh;
      int dd = mm % NB_; int bp2 = mm / NB_; int pp = bp2 % NPIX_; int bb = bp2 / NPIX_;
      out[(((size_t)bb * NCH_ + n) * NPIX_ + pp) * NB_ + dd] = acc[nt][r];
    }
  }
}

// ---------------------------------------------------------------------------
extern "C" void kernel_launch(void* const* d_in, const int* in_sizes, int n_in,
                              void* d_out, int out_size, void* d_ws, size_t ws_size,
                              hipStream_t stream) {
  const float* x     = (const float*)d_in[0];
  const float* W_in  = (const float*)d_in[1];
  const float* b_in  = (const float*)d_in[2];
  const float* W_cs  = (const float*)d_in[3];
  const float* b_cs  = (const float*)d_in[4];
  const float* W_cc  = (const float*)d_in[5];
  const float* b_cc  = (const float*)d_in[6];
  const float* W_xp  = (const float*)d_in[7];
  const float* W_xcp = (const float*)d_in[8];
  const float* W_dt  = (const float*)d_in[9];
  const float* b_dt  = (const float*)d_in[10];
  const float* A_log = (const float*)d_in[11];
  const float* Dvec  = (const float*)d_in[12];
  const float* W_out = (const float*)d_in[13];
  const float* b_out = (const float*)d_in[14];
  float* out = (float*)d_out;

  // workspace layout (floats) — total 6,572,800 f32 = 26.3 MB (fits L2)
  float* ws  = (float*)d_ws;
  float* xr  = ws;                   // (B,128,NPIX,NB)      2,867,200
  float* xc  = xr  + 2867200;        // (B,64,NB)               25,600
  float* Eb  = xc  + 25600;          // (B,16,NB)                6,400
  float* xsb = Eb  + 6400;           // (BN,64,L,NB)         1,433,600
  float* dpb = xsb + 1433600;        // (BN,L,NB,4)             89,600
  float* Bmb = dpb + 89600;          // (BN,L,NB,16)           358,400
  float* Cmb = Bmb + 358400;         // (BN,L,NB,16)           358,400
  float* ymb = Cmb + 358400;         // (B,64,NPIX,NB)       1,433,600

  k_pw_in  <<<175,  256, 0, stream>>>(x, W_in, b_in, xr);                 // 1400 wave-tiles
  k_conv_xc<<<100,  256, 0, stream>>>(xr, W_cc, b_cc, W_xcp, xc, Eb);
  k_conv_xs<<<182,  256, 0, stream>>>(xr, W_cs, b_cs, xsb);               // 1456 wave-tiles
  k_pw_dbc <<<175,  256, 0, stream>>>(xsb, W_xp, dpb, Bmb, Cmb);
  k_scan   <<<1024, 224, 0, stream>>>(xr, xc, Eb, xsb, dpb, Bmb, Cmb,
                                      W_dt, b_dt, A_log, Dvec, ymb);
  k_pw_out <<<175,  256, 0, stream>>>(ymb, W_out, b_out, out);
}